// Self_Attn_loc_90795608637910
// MI455X (gfx1250) — compile-verified
//
#include <hip/hip_runtime.h>
#include <math.h>

// Self-attention-style location softmax:
//   d[i,j]   = Dmat[current[i], history[j]]
//   e[i,j]   = (j <= i) ? 1/d[i,j] : 0
//   out[i,:] = softmax(e[i,:])   (full row; masked entries contribute exp(-max))
//
// MI455X reasoning: no matmul -> WMMA inapplicable; gather/HBM bound
// (~55-80MB first-touch reads, L2-resident working set; 16MB streamed output).
// CDNA5 paths: async global->LDS staging of history (ASYNCcnt +
// s_wait_asynccnt), exec-masked gathers that skip masked column chunks,
// wave32 shuffle reductions, non-temporal output stores.
// VALU trimmed with v_rcp_f32 / v_exp_f32 (1-ulp-class, far inside f32
// validation tolerance; d in [0.5,2], exp args in [-2,0]).

constexpr int SEQ    = 2048;          // history length (= columns)
constexpr int BLOCK  = 256;           // 8 wave32 waves per workgroup
constexpr int PER_T  = SEQ / BLOCK;   // 8 elements per thread
constexpr int NWAVES = BLOCK / 32;    // 8

__global__ __launch_bounds__(BLOCK) void attn_loc_softmax_kernel(
    const int*   __restrict__ history,
    const int*   __restrict__ current,
    const float* __restrict__ Dmat,
    float*       __restrict__ out,
    int npoi)
{
#if defined(__HIP_DEVICE_COMPILE__)
    __shared__ int   hist[SEQ];          // 8 KB
    __shared__ float red_max[NWAVES];
    __shared__ float red_sum[NWAVES];

    const int tid = threadIdx.x;
    const int row = (int)blockIdx.x;

    // ---- Stage history[] (8 KB) into LDS via CDNA5 async global->LDS copies.
    // 256 lanes * 16B * 2 issues = 8 KB. Per CDNA5 ISA, INST_OFFSET
    // is added to BOTH the global and LDS addresses, so one offset:4096
    // handles the second half.
    {
        unsigned long long g =
            (unsigned long long)(const void*)history + (unsigned)(tid * 16);
        // Low 32 bits of a generic pointer to LDS = workgroup-relative LDS
        // byte offset (flat-aperture rule).
        unsigned l =
            (unsigned)(unsigned long long)(void*)&hist[0] + (unsigned)(tid * 16);
        asm volatile(
            "global_load_async_to_lds_b128 %0, %1, off\n\t"
            "global_load_async_to_lds_b128 %0, %1, off offset:4096"
            :: "v"(l), "v"(g) : "memory");
        asm volatile("s_wait_asynccnt 0" ::: "memory");
    }
    __syncthreads();

    // Uniform per block -> scalar load + scalar address math.
    const int c = current[row];
    const float* Drow = Dmat + (size_t)c * (size_t)npoi;

    // ---- Pass 1: masked gather + fast reciprocal, track row max (all >= 0).
    // Keeping the gather under the mask lets fully-masked chunks be skipped
    // (s_cbranch_execz), minimizing first-touch HBM reads of distance rows.
    float e[PER_T];
    float m = 0.0f;
    #pragma unroll
    for (int k = 0; k < PER_T; ++k) {
        const int j = tid + k * BLOCK;
        float ev = 0.0f;
        if (j <= row) {
            const int h = hist[j];                    // LDS, conflict-free
            ev = __builtin_amdgcn_rcpf(Drow[h]);      // v_rcp_f32, ~1 ulp
        }
        e[k] = ev;
        m = fmaxf(m, ev);
    }

    // ---- Row max: wave32 shuffle tree, then 8-way cross-wave via LDS.
    #pragma unroll
    for (int off = 16; off > 0; off >>= 1)
        m = fmaxf(m, __shfl_xor(m, off, 32));
    if ((tid & 31) == 0) red_max[tid >> 5] = m;
    __syncthreads();
    m = red_max[0];
    #pragma unroll
    for (int w = 1; w < NWAVES; ++w) m = fmaxf(m, red_max[w]);

    // ---- Pass 2: exp(e - m) for ALL entries (masked ones are exp(-m)), sum.
    float s = 0.0f;
    #pragma unroll
    for (int k = 0; k < PER_T; ++k) {
        const float x = __expf(e[k] - m);    // v_exp_f32 path, args in [-2,0]
        e[k] = x;
        s += x;
    }
    #pragma unroll
    for (int off = 16; off > 0; off >>= 1)
        s += __shfl_xor(s, off, 32);
    if ((tid & 31) == 0) red_sum[tid >> 5] = s;
    __syncthreads();
    s = red_sum[0];
    #pragma unroll
    for (int w = 1; w < NWAVES; ++w) s += red_sum[w];

    // ---- Normalize + streamed (non-temporal) store: output is write-once;
    // keep L2 for the distance-row working set instead.
    const float inv = __builtin_amdgcn_rcpf(s);
    const size_t base = (size_t)row * (size_t)SEQ;
    #pragma unroll
    for (int k = 0; k < PER_T; ++k) {
        const int j = tid + k * BLOCK;
        __builtin_nontemporal_store(e[k] * inv, out + base + j);
    }
#endif // __HIP_DEVICE_COMPILE__
}

extern "C" void kernel_launch(void* const* d_in, const int* in_sizes, int n_in,
                              void* d_out, int out_size, void* d_ws, size_t ws_size,
                              hipStream_t stream) {
    const int*   history = (const int*)d_in[0];   // (2048,) int32
    const int*   current = (const int*)d_in[1];   // (2048,) int32
    const float* Dmat    = (const float*)d_in[2]; // (10000,10000) f32
    float*       out     = (float*)d_out;         // (2048,2048) f32

    const int state = in_sizes[1];                               // rows = 2048
    const int npoi  = (int)(sqrt((double)in_sizes[2]) + 0.5);    // 10000

    attn_loc_softmax_kernel<<<dim3(state), dim3(BLOCK), 0, stream>>>(
        history, current, Dmat, out, npoi);
    (void)n_in; (void)out_size; (void)d_ws; (void)ws_size;
}